// CRF_55688545960122
// MI455X (gfx1250) — compile-verified
//
#include <hip/hip_runtime.h>
#include <stddef.h>

// CRF forward algorithm on MI455X (gfx1250, wave32).
//
// Reformulated as a scaled positive linear recurrence so the inner loop is a
// fp32 WMMA GEMM:   y[b,n] = sum_p W[b,p] * expT[p,n] * exp(feat[t,b,n])
//                   W'     = y / rowmax(y);   s[b] += log(rowmax)
// answer[b] = s[b] + log(sum_n W'[b,n])
//
// One wave handles 16 batches (WMMA N... here M=batch, N=next-tag).
// K (=50 tags) padded to 64: 4 N-tiles of 16, 16 k-chunks of 4 (fp32 WMMA is
// 16x16x4).  64 v_wmma_f32_16x16x4_f32 per timestep per wave, 4 independent
// accumulator chains.  exp(transitions)^T is register-resident (128 VGPRs).

typedef __attribute__((ext_vector_type(2))) float v2f;
typedef __attribute__((ext_vector_type(8))) float v8f;

#define TT 1024
#define KK 50
#define BB 512
#define LSTRIDE 66   // LDS row stride in floats: even (b64-aligned) and != 64 (bank spread)

__launch_bounds__(32, 1)
__global__ void crf_fwd_wmma(const float* __restrict__ feats,
                             const float* __restrict__ trans,
                             float* __restrict__ out) {
  __shared__ float lw[16 * LSTRIDE];   // W' round-trip: [batch16][prev64 (+pad)]

  const int lane = threadIdx.x & 31;
  const int half = lane >> 4;          // 0: lanes 0-15, 1: lanes 16-31
  const int l16  = lane & 15;
  const int b0   = blockIdx.x * 16;    // this wave's batch group

  // ---- preload Et = exp(transitions)^T in WMMA B layout (4x16 f32 tiles) ----
  // B operand: VGPR v, lane l -> row K(prev-in-chunk) = v + 2*half, col N(next) = l16
  v2f Et[16][4];
#pragma unroll
  for (int c = 0; c < 16; ++c)
#pragma unroll
    for (int tl = 0; tl < 4; ++tl)
#pragma unroll
      for (int v = 0; v < 2; ++v) {
        int prev = 4 * c + v + 2 * half;
        int next = 16 * tl + l16;
        float e = 0.0f;
        if (prev < KK && next < KK) e = __expf(trans[next * KK + prev]);
        Et[c][tl][v] = e;
      }

  // ---- init W in WMMA A layout (16x4 f32): M(batch)=l16, K(prev)=v+2*half ----
  // fv0 = [0, -1e4, ...]  ->  w0 = [1, 0, ..., 0], scale s0 = 0
  v2f W[16];
#pragma unroll
  for (int c = 0; c < 16; ++c) { W[c][0] = 0.0f; W[c][1] = 0.0f; }
  if (half == 0) W[0][0] = 1.0f;       // prev == 0

  float s[8];
#pragma unroll
  for (int r = 0; r < 8; ++r) s[r] = 0.0f;

  // y-layout (C/D): VGPR r, lane l -> batch = r + 8*half, next = 16*tl + l16
  const float* fbase[8];
#pragma unroll
  for (int r = 0; r < 8; ++r)
    fbase[r] = feats + (size_t)(b0 + r + 8 * half) * (TT * KK) + l16;

  float y[4][8];
  float rmax[8];

  for (int t = 0; t < TT; ++t) {
    // ---- emission scores for this step (coalesced 64B per half-wave) ----
    float f[4][8];
#pragma unroll
    for (int tl = 0; tl < 4; ++tl) {
      const bool nvalid = (16 * tl + l16) < KK;   // pad tags get multiplier 0
#pragma unroll
      for (int r = 0; r < 8; ++r) {
        float fr = 0.0f;
        if (nvalid) fr = fbase[r][(size_t)t * KK + 16 * tl];
        f[tl][r] = fr;
      }
    }

    // ---- y = W · Et  (4 tiles x 16 k-chunks of v_wmma_f32_16x16x4_f32) ----
    v8f acc[4];
#pragma unroll
    for (int tl = 0; tl < 4; ++tl) {
      v8f a = {};
#pragma unroll
      for (int c = 0; c < 16; ++c)
        a = __builtin_amdgcn_wmma_f32_16x16x4_f32(
            /*neg_a=*/false, W[c], /*neg_b=*/false, Et[c][tl],
            /*c_mod=*/(short)0, a, /*reuse_a=*/false, /*reuse_b=*/false);
      acc[tl] = a;
    }

    // ---- y *= exp(feat); per-batch max over the 64 next-tags ----
#pragma unroll
    for (int tl = 0; tl < 4; ++tl) {
      const bool nvalid = (16 * tl + l16) < KK;
#pragma unroll
      for (int r = 0; r < 8; ++r)
        y[tl][r] = nvalid ? acc[tl][r] * __expf(f[tl][r]) : 0.0f;
    }
#pragma unroll
    for (int r = 0; r < 8; ++r) {
      float m = fmaxf(fmaxf(y[0][r], y[1][r]), fmaxf(y[2][r], y[3][r]));
      m = fmaxf(m, __shfl_xor(m, 1, 32));
      m = fmaxf(m, __shfl_xor(m, 2, 32));
      m = fmaxf(m, __shfl_xor(m, 4, 32));
      m = fmaxf(m, __shfl_xor(m, 8, 32));   // stays inside the owning 16-lane half
      rmax[r] = m;
      s[r] += __logf(m);
    }

    // ---- normalize and transpose C-layout -> A-layout via LDS ----
#pragma unroll
    for (int r = 0; r < 8; ++r) {
      const float inv = __builtin_amdgcn_rcpf(rmax[r]);
#pragma unroll
      for (int tl = 0; tl < 4; ++tl)
        lw[(r + 8 * half) * LSTRIDE + 16 * tl + l16] = y[tl][r] * inv;
    }
    __syncthreads();                       // single wave: cheap (ds waits)
#pragma unroll
    for (int c = 0; c < 16; ++c)
      W[c] = *(const v2f*)&lw[l16 * LSTRIDE + 4 * c + 2 * half];
    __syncthreads();
  }

  // ---- final logsumexp: s + log(sum_n y/rmax) using last iteration's y ----
  float rs[8];
#pragma unroll
  for (int r = 0; r < 8; ++r) {
    float t0 = (y[0][r] + y[1][r]) + (y[2][r] + y[3][r]);
    t0 += __shfl_xor(t0, 1, 32);
    t0 += __shfl_xor(t0, 2, 32);
    t0 += __shfl_xor(t0, 4, 32);
    t0 += __shfl_xor(t0, 8, 32);
    rs[r] = t0;
  }
  if (l16 == 0) {
#pragma unroll
    for (int r = 0; r < 8; ++r)
      out[b0 + r + 8 * half] = s[r] + __logf(rs[r]) - __logf(rmax[r]);
  }
}

extern "C" void kernel_launch(void* const* d_in, const int* in_sizes, int n_in,
                              void* d_out, int out_size, void* d_ws, size_t ws_size,
                              hipStream_t stream) {
  (void)in_sizes; (void)n_in; (void)out_size; (void)d_ws; (void)ws_size;
  const float* feats = (const float*)d_in[0];   // [512,1024,50] fp32
  const float* trans = (const float*)d_in[1];   // [50,50] fp32
  float* out = (float*)d_out;                   // [512] fp32
  crf_fwd_wmma<<<dim3(BB / 16), dim3(32), 0, stream>>>(feats, trans, out);
}